// AudioMambaModel_32865089749430
// MI455X (gfx1250) — compile-verified
//
#include <hip/hip_runtime.h>
#include <cstdint>
#include <cstddef>

// ---------------- model dims ----------------
#define BB     8
#define NMELS  128
#define TLEN   1024
#define DD     768
#define EE     1536
#define E2     3072          // 2*E
#define RR     48
#define NN     16
#define KK     4
#define LL     513           // 512 patches + cls
#define MIDP   256
#define NCLS   10
#define NDEPTH 12
#define MTOT   (BB*LL)       // 4104 rows
#define NPATCH 4096          // B * 512
#define PATCHK 256           // 16*16

typedef float v2f __attribute__((ext_vector_type(2)));
typedef float v8f __attribute__((ext_vector_type(8)));

__device__ __forceinline__ float softplusf(float x) {
    return x > 20.f ? x : log1pf(__expf(x));
}
__device__ __forceinline__ float siluf(float x) {
    return x / (1.f + __expf(-x));
}

// ---------------------------------------------------------------------------
// fp32 WMMA GEMM:  C[M x Nout] = (A (+A2)) [M x K, row stride lda] * W[Nout x K]^T
// One wave computes a (MT*16) x (NT*16) tile via V_WMMA_F32_16X16X4_F32.
// M-blocking (MT=2) reuses each B (weight) fragment MT times -> halves the
// weight-stream bytes per WMMA. HASA2 templated so the residual-combine GEMM
// has a branch-free inner loop (no per-iteration EXEC save/restore).
// epi: 0 = store, 1 = softplus(store) [after bias], 2 = accumulate into C.
// ---------------------------------------------------------------------------
template<int MT, int NT, bool HASA2>
__global__ void gemm_wmma_f32(const float* __restrict__ A,
                              const float* __restrict__ A2,
                              const float* __restrict__ W,
                              const float* __restrict__ bias,
                              float* __restrict__ C,
                              int M, int Kd, int Nout, int lda, int epi)
{
    const int lane  = threadIdx.x & 31;
    const int mBase = blockIdx.x * MT;        // first 16-row tile of this wave
    const int nStrip = blockIdx.y;            // NT*16 cols of C
    const int k0    = (lane >> 4) << 1;       // 0 or 2
    const int nBase = nStrip * (NT * 16) + (lane & 15);

    const float* arow[MT];
    const float* arow2[MT];
#pragma unroll
    for (int mt = 0; mt < MT; ++mt) {
        const int mld = (mBase + mt) * 16 + (lane & 15);
        const int mClamp = mld < M ? mld : (M - 1);    // clamp edge rows (stores guarded)
        arow[mt]  = A + (size_t)mClamp * lda;
        arow2[mt] = HASA2 ? (A2 + (size_t)mClamp * lda) : nullptr;
    }

    v8f zero = {};
    v8f acc[MT][NT];
#pragma unroll
    for (int mt = 0; mt < MT; ++mt)
#pragma unroll
        for (int t = 0; t < NT; ++t) acc[mt][t] = zero;

    // uniform trip count -> EXEC stays all-ones across every WMMA
    for (int kk = 0; kk < Kd; kk += 4) {
        const int k = kk + k0;
        v2f a[MT];
#pragma unroll
        for (int mt = 0; mt < MT; ++mt) {
            a[mt].x = arow[mt][k];
            a[mt].y = arow[mt][k + 1];
            if (HASA2) { a[mt].x += arow2[mt][k]; a[mt].y += arow2[mt][k + 1]; }
        }
#pragma unroll
        for (int t = 0; t < NT; ++t) {
            const float* wrow = W + (size_t)(nBase + t * 16) * Kd;
            v2f bv;
            bv.x = wrow[k];
            bv.y = wrow[k + 1];
#pragma unroll
            for (int mt = 0; mt < MT; ++mt) {
                acc[mt][t] = __builtin_amdgcn_wmma_f32_16x16x4_f32(
                    false, a[mt], false, bv, (short)0, acc[mt][t], false, false);
            }
        }
    }

#pragma unroll
    for (int mt = 0; mt < MT; ++mt) {
        const int mrow = (mBase + mt) * 16 + ((lane >> 4) << 3); // D VGPR i -> row mrow + i
#pragma unroll
        for (int t = 0; t < NT; ++t) {
            const int n = nStrip * (NT * 16) + t * 16 + (lane & 15);
            const float badd = bias ? bias[n] : 0.f;
#pragma unroll
            for (int i = 0; i < 8; ++i) {
                const int mm = mrow + i;
                if (mm < M) {
                    float v = acc[mt][t][i] + badd;
                    if (epi == 1) v = softplusf(v);
                    float* dst = C + (size_t)mm * Nout + n;
                    if (epi == 2) *dst += v; else *dst = v;
                }
            }
        }
    }
}

// ---------------------------------------------------------------------------
// Patch gather: Ap[m][k] = x[b, prow*16 + k/16, pcol*16 + k%16]
// m = b*512 + prow*64 + pcol, k = r*16 + c  (matches patch_w flat layout)
// ---------------------------------------------------------------------------
__global__ void patch_gather(const float* __restrict__ x, float* __restrict__ Ap)
{
    const size_t idx = (size_t)blockIdx.x * blockDim.x + threadIdx.x;
    if (idx >= (size_t)NPATCH * PATCHK) return;
    const int k = (int)(idx & 255);
    const int m = (int)(idx >> 8);
    const int b = m >> 9;
    const int p = m & 511;
    const int prow = p >> 6, pcol = p & 63;
    const int r = k >> 4, c = k & 15;
    Ap[idx] = x[((size_t)b * NMELS + prow * 16 + r) * TLEN + pcol * 16 + c];
}

// h[b,l,:] = cls (l==MID) else tok[b, l<MID? l : l-1, :] + patch_b
__global__ void assemble_h(const float* __restrict__ tok, const float* __restrict__ cls,
                           const float* __restrict__ pb, float* __restrict__ h)
{
    const size_t idx = (size_t)blockIdx.x * blockDim.x + threadIdx.x;
    if (idx >= (size_t)MTOT * DD) return;
    const int d = (int)(idx % DD);
    const int l = (int)((idx / DD) % LL);
    const int b = (int)(idx / ((size_t)DD * LL));
    float v;
    if (l == MIDP) {
        v = cls[d];
    } else {
        const int p = (l < MIDP) ? l : (l - 1);
        v = tok[((size_t)b * 512 + p) * DD + d] + pb[d];
    }
    h[idx] = v;
}

// ---------------------------------------------------------------------------
// LayerNorm: one wave per row of [nrows x dim], shuffle reduction (wave32)
// ---------------------------------------------------------------------------
__global__ void layernorm_rows(const float* __restrict__ in, const float* __restrict__ w,
                               const float* __restrict__ bvec, float* __restrict__ out,
                               int nrows, int dim)
{
    const int wid  = (int)((blockIdx.x * blockDim.x + threadIdx.x) >> 5);
    const int lane = threadIdx.x & 31;
    if (wid >= nrows) return;
    const float* row = in + (size_t)wid * dim;
    float s = 0.f, s2 = 0.f;
    for (int i = lane; i < dim; i += 32) { float v = row[i]; s += v; s2 += v * v; }
#pragma unroll
    for (int o = 16; o; o >>= 1) { s += __shfl_xor(s, o, 32); s2 += __shfl_xor(s2, o, 32); }
    const float mean = s / dim;
    const float inv  = rsqrtf(s2 / dim - mean * mean + 1e-5f);
    for (int i = lane; i < dim; i += 32)
        out[(size_t)wid * dim + i] = (row[i] - mean) * inv * w[i] + bvec[i];
}

// ---------------------------------------------------------------------------
// Causal depthwise conv (K=4) + SiLU. Reads x-half of xz (row stride 2E).
// rev=1: operate on the time-reversed sequence (output stays in reversed frame).
// ---------------------------------------------------------------------------
__global__ void dwconv_silu(const float* __restrict__ xz, const float* __restrict__ w,
                            const float* __restrict__ bias, float* __restrict__ out, int rev)
{
    const size_t idx = (size_t)blockIdx.x * blockDim.x + threadIdx.x;
    if (idx >= (size_t)MTOT * EE) return;
    const int e = (int)(idx % EE);
    const int l = (int)((idx / EE) % LL);
    const int b = (int)(idx / ((size_t)EE * LL));
    float acc = bias[e];
#pragma unroll
    for (int k = 0; k < KK; ++k) {
        const int ll = l - (KK - 1) + k;
        if (ll >= 0) {
            const int src = rev ? (LL - 1 - ll) : ll;
            acc += w[e * KK + k] * xz[((size_t)b * LL + src) * E2 + e];
        }
    }
    out[idx] = siluf(acc);
}

// ---------------------------------------------------------------------------
// Selective-scan: thread = one (b, e) channel, 16 states in registers.
// dbc row = [dt(48) | B(16) | C(16)], shared across e -> float4 broadcast loads.
// Output written at forward-frame index lo (= L-1-l for rev) with silu(z) gate.
// ---------------------------------------------------------------------------
__global__ void ssm_scan(const float* __restrict__ dbc, const float* __restrict__ dlt,
                         const float* __restrict__ u,   const float* __restrict__ xz,
                         const float* __restrict__ A_log, const float* __restrict__ Dvec,
                         float* __restrict__ y, int rev)
{
    const int e = blockIdx.x * blockDim.x + threadIdx.x;
    const int b = blockIdx.y;
    float Ae[NN], h[NN];
#pragma unroll
    for (int n = 0; n < NN; ++n) {
        Ae[n] = -__expf(A_log[(size_t)e * NN + n]);
        h[n]  = 0.f;
    }
    const float De = Dvec[e];
    for (int l = 0; l < LL; ++l) {
        const size_t row = (size_t)b * LL + l;
        const float dt = dlt[row * EE + e];
        const float ut = u[row * EE + e];
        const float du = dt * ut;
        const float4* Bv = (const float4*)(dbc + row * 80 + RR);
        const float4* Cv = (const float4*)(dbc + row * 80 + RR + NN);
        float yv = 0.f;
#pragma unroll
        for (int q = 0; q < 4; ++q) {
            const float4 Bq = Bv[q];
            const float4 Cq = Cv[q];
            h[4*q+0] = h[4*q+0] * __expf(dt * Ae[4*q+0]) + du * Bq.x; yv += h[4*q+0] * Cq.x;
            h[4*q+1] = h[4*q+1] * __expf(dt * Ae[4*q+1]) + du * Bq.y; yv += h[4*q+1] * Cq.y;
            h[4*q+2] = h[4*q+2] * __expf(dt * Ae[4*q+2]) + du * Bq.z; yv += h[4*q+2] * Cq.z;
            h[4*q+3] = h[4*q+3] * __expf(dt * Ae[4*q+3]) + du * Bq.w; yv += h[4*q+3] * Cq.w;
        }
        const int lo = rev ? (LL - 1 - l) : l;
        const size_t orow = (size_t)b * LL + lo;
        const float z = xz[orow * E2 + EE + e];
        y[orow * EE + e] = (yv + ut * De) * siluf(z);
    }
}

// ---------------------------------------------------------------------------
// Head: only row MID of h needs the final LN. One wave per batch element:
// LN(fnorm) -> LN(head_norm) -> 10 dot products. All in registers (768/32=24).
// ---------------------------------------------------------------------------
__global__ void head_kernel(const float* __restrict__ h,
                            const float* __restrict__ fw, const float* __restrict__ fb,
                            const float* __restrict__ hw, const float* __restrict__ hb,
                            const float* __restrict__ pw, const float* __restrict__ pb,
                            float* __restrict__ out)
{
    const int b = blockIdx.x;
    const int lane = threadIdx.x;
    const float* row = h + ((size_t)b * LL + MIDP) * DD;
    float v[24];
    float s = 0.f, s2 = 0.f;
#pragma unroll
    for (int i = 0; i < 24; ++i) { float t = row[lane + 32 * i]; v[i] = t; s += t; s2 += t * t; }
#pragma unroll
    for (int o = 16; o; o >>= 1) { s += __shfl_xor(s, o, 32); s2 += __shfl_xor(s2, o, 32); }
    float mean = s * (1.f / DD);
    float inv  = rsqrtf(s2 * (1.f / DD) - mean * mean + 1e-5f);
    s = 0.f; s2 = 0.f;
#pragma unroll
    for (int i = 0; i < 24; ++i) {
        const int d = lane + 32 * i;
        float t = (v[i] - mean) * inv * fw[d] + fb[d];
        v[i] = t; s += t; s2 += t * t;
    }
#pragma unroll
    for (int o = 16; o; o >>= 1) { s += __shfl_xor(s, o, 32); s2 += __shfl_xor(s2, o, 32); }
    mean = s * (1.f / DD);
    inv  = rsqrtf(s2 * (1.f / DD) - mean * mean + 1e-5f);
#pragma unroll
    for (int i = 0; i < 24; ++i) {
        const int d = lane + 32 * i;
        v[i] = (v[i] - mean) * inv * hw[d] + hb[d];
    }
    for (int c = 0; c < NCLS; ++c) {
        float p = 0.f;
#pragma unroll
        for (int i = 0; i < 24; ++i) { const int d = lane + 32 * i; p += v[i] * pw[c * DD + d]; }
#pragma unroll
        for (int o = 16; o; o >>= 1) p += __shfl_xor(p, o, 32);
        if (lane == 0) out[b * NCLS + c] = p + pb[c];
    }
}

// ---------------------------------------------------------------------------
extern "C" void kernel_launch(void* const* d_in, const int* in_sizes, int n_in,
                              void* d_out, int out_size, void* d_ws, size_t ws_size,
                              hipStream_t stream)
{
    (void)in_sizes; (void)n_in; (void)out_size; (void)ws_size;
    // setup_inputs() dict order
    const float* x           = (const float*)d_in[0];
    const float* patch_w     = (const float*)d_in[1];
    const float* patch_b     = (const float*)d_in[2];
    const float* cls_tok     = (const float*)d_in[3];
    const float* norm_w      = (const float*)d_in[4];
    const float* norm_b      = (const float*)d_in[5];
    const float* in_proj_w   = (const float*)d_in[6];
    const float* conv_w_f    = (const float*)d_in[7];
    const float* conv_b_f    = (const float*)d_in[8];
    const float* x_proj_w_f  = (const float*)d_in[9];
    const float* dt_proj_w_f = (const float*)d_in[10];
    const float* dt_proj_b_f = (const float*)d_in[11];
    const float* A_log_f     = (const float*)d_in[12];
    const float* D_f         = (const float*)d_in[13];
    const float* conv_w_b    = (const float*)d_in[14];
    const float* conv_b_b    = (const float*)d_in[15];
    const float* x_proj_w_b  = (const float*)d_in[16];
    const float* dt_proj_w_b = (const float*)d_in[17];
    const float* dt_proj_b_b = (const float*)d_in[18];
    const float* A_log_b     = (const float*)d_in[19];
    const float* D_b         = (const float*)d_in[20];
    const float* out_proj_w  = (const float*)d_in[21];
    const float* fnorm_w     = (const float*)d_in[22];
    const float* fnorm_b     = (const float*)d_in[23];
    const float* head_norm_w = (const float*)d_in[24];
    const float* head_norm_b = (const float*)d_in[25];
    const float* head_w      = (const float*)d_in[26];
    const float* head_b      = (const float*)d_in[27];

    // ---- workspace carve-up (floats, 256B-aligned chunks), ~180 MB total ----
    float* ws = (float*)d_ws;
    size_t off = 0;
    auto alloc = [&](size_t n) { float* p = ws + off; off += (n + 63) & ~(size_t)63; return p; };
    float* hbuf = alloc((size_t)MTOT * DD);   // residual stream
    float* hn   = alloc((size_t)MTOT * DD);   // normed h
    float* xzb  = alloc((size_t)MTOT * E2);   // in_proj output [x | z]
    float* xcf  = alloc((size_t)MTOT * EE);   // fwd conv out (u_f), later reused as y_b
    float* xcb  = alloc((size_t)MTOT * EE);   // bwd conv out (u_b)
    float* dbcf = alloc((size_t)MTOT * 80);   // fwd [dt|B|C]
    float* dbcb = alloc((size_t)MTOT * 80);   // bwd [dt|B|C]
    float* dltf = alloc((size_t)MTOT * EE);   // fwd delta, scan writes y_f in place
    float* dltb = alloc((size_t)MTOT * EE);   // bwd delta
    // patch-phase buffers alias xzb (used strictly before xz is written)
    float* Ap  = xzb;                           // 4096 x 256
    float* tok = xzb + (size_t)NPATCH * PATCHK; // 4096 x 768

    // ---- patch embed as WMMA GEMM ----
    patch_gather<<<(NPATCH * PATCHK + 255) / 256, 256, 0, stream>>>(x, Ap);
    gemm_wmma_f32<2, 4, false><<<dim3(NPATCH / 32, DD / 64), 32, 0, stream>>>(
        Ap, nullptr, patch_w, nullptr, tok, NPATCH, PATCHK, DD, PATCHK, 0);
    assemble_h<<<((size_t)MTOT * DD + 255) / 256, 256, 0, stream>>>(tok, cls_tok, patch_b, hbuf);

    const int MB = (MTOT + 31) / 32;          // 129 blocks of 2 row-tiles
    const size_t nelE = (size_t)MTOT * EE;

    for (int i = 0; i < NDEPTH; ++i) {
        // LN
        layernorm_rows<<<(MTOT + 7) / 8, 256, 0, stream>>>(
            hbuf, norm_w + i * DD, norm_b + i * DD, hn, MTOT, DD);
        // in_proj: [M x 768] x [3072 x 768]^T
        gemm_wmma_f32<2, 4, false><<<dim3(MB, E2 / 64), 32, 0, stream>>>(
            hn, nullptr, in_proj_w + (size_t)i * E2 * DD, nullptr, xzb, MTOT, DD, E2, DD, 0);
        // depthwise conv + silu (fwd, bwd)
        dwconv_silu<<<(nelE + 255) / 256, 256, 0, stream>>>(
            xzb, conv_w_f + (size_t)i * EE * KK, conv_b_f + i * EE, xcf, 0);
        dwconv_silu<<<(nelE + 255) / 256, 256, 0, stream>>>(
            xzb, conv_w_b + (size_t)i * EE * KK, conv_b_b + i * EE, xcb, 1);
        // x_proj: [M x 1536] x [80 x 1536]^T
        gemm_wmma_f32<2, 1, false><<<dim3(MB, 5), 32, 0, stream>>>(
            xcf, nullptr, x_proj_w_f + (size_t)i * 80 * EE, nullptr, dbcf, MTOT, EE, 80, EE, 0);
        gemm_wmma_f32<2, 1, false><<<dim3(MB, 5), 32, 0, stream>>>(
            xcb, nullptr, x_proj_w_b + (size_t)i * 80 * EE, nullptr, dbcb, MTOT, EE, 80, EE, 0);
        // dt_proj + bias + softplus: [M x 48 (lda=80)] x [1536 x 48]^T
        gemm_wmma_f32<2, 4, false><<<dim3(MB, EE / 64), 32, 0, stream>>>(
            dbcf, nullptr, dt_proj_w_f + (size_t)i * EE * RR, dt_proj_b_f + i * EE,
            dltf, MTOT, RR, EE, 80, 1);
        gemm_wmma_f32<2, 4, false><<<dim3(MB, EE / 64), 32, 0, stream>>>(
            dbcb, nullptr, dt_proj_w_b + (size_t)i * EE * RR, dt_proj_b_b + i * EE,
            dltb, MTOT, RR, EE, 80, 1);
        // scans: fwd writes y_f over dltf (in place); bwd writes y_b over xcf
        ssm_scan<<<dim3(EE / 256, BB), 256, 0, stream>>>(
            dbcf, dltf, xcf, xzb, A_log_f + (size_t)i * EE * NN, D_f + i * EE, dltf, 0);
        ssm_scan<<<dim3(EE / 256, BB), 256, 0, stream>>>(
            dbcb, dltb, xcb, xzb, A_log_b + (size_t)i * EE * NN, D_b + i * EE, xcf, 1);
        // out_proj on (y_f + y_b), accumulate residual into h
        gemm_wmma_f32<2, 4, true><<<dim3(MB, DD / 64), 32, 0, stream>>>(
            dltf, xcf, out_proj_w + (size_t)i * DD * EE, nullptr, hbuf, MTOT, EE, DD, EE, 2);
    }

    head_kernel<<<BB, 32, 0, stream>>>(hbuf, fnorm_w, fnorm_b, head_norm_w, head_norm_b,
                                       head_w, head_b, (float*)d_out);
}